// GAT_46712064311583
// MI455X (gfx1250) — compile-verified
//
#include <hip/hip_runtime.h>
#include <math.h>

// ---------------------------------------------------------------------------
// GAT 2-layer forward for MI455X (gfx1250, wave32).
//  - GEMMs use V_WMMA_F32_16X16X4_F32 (exact fp32; GEMMs are compute-trivial
//    here, the workload is bound by ~1.7 GB of edge gather/scatter traffic).
//  - Edge softmax uses ordered-int atomicMax + f32 atomicAdd.
// ---------------------------------------------------------------------------

typedef __attribute__((ext_vector_type(2))) float v2f;
typedef __attribute__((ext_vector_type(8))) float v8f;

#define KDIM 256
#define NEG_SLOPE 0.2f

// Monotonic float<->int mapping so integer atomicMax orders like float max.
__device__ __forceinline__ int f2ord(float f) {
  int i = __float_as_int(f);
  return i >= 0 ? i : (i ^ 0x7fffffff);
}
__device__ __forceinline__ float ord2f(int i) {
  return __int_as_float(i >= 0 ? i : (i ^ 0x7fffffff));
}

// ---------------------------------------------------------------------------
// Generic fp32 WMMA GEMM: C[M,Ncols] = A[M,256] * B[256,Ncols].
// One wave computes one 16x16 output tile; A tile (16x256) staged in LDS.
// grid.x = ceil(M/16), grid.y covers column-tile groups of (blockDim/32).
// Inner loop is branch-free: out-of-range columns are clamped (loads stay
// in-bounds, results discarded at the masked store), so EXEC stays all-1s.
// ---------------------------------------------------------------------------
__global__ __launch_bounds__(256) void k_gemm_wmma(
    const float* __restrict__ A, const float* __restrict__ B,
    float* __restrict__ C, int M, int Ncols) {
  __shared__ float As[16 * KDIM];

  const int m0 = blockIdx.x * 16;

  // Cooperative, coalesced load of the 16x256 A tile.
  for (int idx = threadIdx.x; idx < 16 * KDIM; idx += blockDim.x) {
    const int r = idx >> 8;
    const int c = idx & (KDIM - 1);
    As[idx] = (m0 + r < M) ? A[(size_t)(m0 + r) * KDIM + c] : 0.0f;
  }
  __syncthreads();

  const int lane = threadIdx.x & 31;
  const int wave = threadIdx.x >> 5;
  const int wavesPerBlock = blockDim.x >> 5;
  const int nTiles = (Ncols + 15) >> 4;
  const int ntile = blockIdx.y * wavesPerBlock + wave;
  if (ntile >= nTiles) return;

  const int n0 = ntile * 16;
  const int mrow = lane & 15;
  const int khalf = (lane >> 4) << 1;        // 0 for lanes 0-15, 2 for 16-31
  const int ncol = n0 + (lane & 15);
  const bool ncol_ok = (ncol < Ncols);
  const int ncolc = ncol_ok ? ncol : (Ncols - 1);  // clamp: keep loads legal

  const float* Ap = &As[mrow * KDIM + khalf];        // LDS, b64 per step
  const float* Bp = B + (size_t)khalf * Ncols + ncolc;

  v8f acc = {};
#pragma unroll 4
  for (int k0 = 0; k0 < KDIM; k0 += 4) {
    // A fragment per ISA layout: VGPR0 = K{0,2}, VGPR1 = K{1,3} across halves.
    v2f a;
    a.x = Ap[k0];
    a.y = Ap[k0 + 1];
    v2f b;
    b.x = Bp[0];
    b.y = Bp[Ncols];
    Bp += (size_t)Ncols * 4;
    acc = __builtin_amdgcn_wmma_f32_16x16x4_f32(
        /*neg_a=*/false, a, /*neg_b=*/false, b,
        /*c_mod=*/(short)0, acc, /*reuse_a=*/false, /*reuse_b=*/false);
  }

  // C layout: VGPR r -> row r (lanes 0-15) / row r+8 (lanes 16-31).
  const int rowbase = m0 + ((lane >> 4) << 3);
  float* Cp = C + (size_t)rowbase * Ncols + ncol;
  if (ncol_ok) {
    if (m0 + 16 <= M) {
      // Wave-uniform fast path: full tile, unconditional 8-row store.
#pragma unroll
      for (int r = 0; r < 8; ++r) Cp[(size_t)r * Ncols] = acc[r];
    } else {
#pragma unroll
      for (int r = 0; r < 8; ++r)
        if (rowbase + r < M) Cp[(size_t)r * Ncols] = acc[r];
    }
  }
}

// ---------------------------------------------------------------------------
// Layer-1 attention scores: el/er[n,h] = sum_d feat[n,h*32+d]*attn[h*32+d].
// One wave per node; lane owns 8 consecutive features (one head per 4 lanes).
// ---------------------------------------------------------------------------
__global__ __launch_bounds__(256) void k_scores1(
    const float* __restrict__ feat, const float* __restrict__ attn_l,
    const float* __restrict__ attn_r, float* __restrict__ el,
    float* __restrict__ er, int N) {
  const int node = blockIdx.x * (blockDim.x >> 5) + (threadIdx.x >> 5);
  if (node >= N) return;
  const int lane = threadIdx.x & 31;

  const float4* fr = (const float4*)(feat + (size_t)node * KDIM);
  const float4* al = (const float4*)attn_l;
  const float4* ar = (const float4*)attn_r;
  float4 f0 = fr[lane * 2], f1 = fr[lane * 2 + 1];
  float4 l0 = al[lane * 2], l1 = al[lane * 2 + 1];
  float4 r0 = ar[lane * 2], r1 = ar[lane * 2 + 1];

  float sl = f0.x * l0.x + f0.y * l0.y + f0.z * l0.z + f0.w * l0.w +
             f1.x * l1.x + f1.y * l1.y + f1.z * l1.z + f1.w * l1.w;
  float sr = f0.x * r0.x + f0.y * r0.y + f0.z * r0.z + f0.w * r0.w +
             f1.x * r1.x + f1.y * r1.y + f1.z * r1.z + f1.w * r1.w;

  // Reduce across the 4 lanes of each head group.
  sl += __shfl_xor(sl, 1, 32); sl += __shfl_xor(sl, 2, 32);
  sr += __shfl_xor(sr, 1, 32); sr += __shfl_xor(sr, 2, 32);
  if ((lane & 3) == 0) {
    const int h = lane >> 2;
    el[node * 8 + h] = sl;
    er[node * 8 + h] = sr;
  }
}

// Layer-2 scores (H=1, D=40): one thread per node.
__global__ __launch_bounds__(256) void k_scores2(
    const float* __restrict__ feat, const float* __restrict__ al,
    const float* __restrict__ ar, float* __restrict__ el,
    float* __restrict__ er, int N) {
  const int n = blockIdx.x * blockDim.x + threadIdx.x;
  if (n >= N) return;
  const float* f = feat + (size_t)n * 40;
  float sl = 0.f, sr = 0.f;
#pragma unroll
  for (int d = 0; d < 40; ++d) {
    const float v = f[d];
    sl += v * al[d];
    sr += v * ar[d];
  }
  el[n] = sl;
  er[n] = sr;
}

// ---------------------------------------------------------------------------
// Edge pass 1: score = leaky_relu(el[src]+er[dst]); segment-max via atomicMax.
// ---------------------------------------------------------------------------
template <int H>
__global__ __launch_bounds__(256) void k_edge_score_max(
    const int* __restrict__ src, const int* __restrict__ dst,
    const float* __restrict__ el, const float* __restrict__ er,
    float* __restrict__ s, int* __restrict__ m, int EH) {
  const int t = blockIdx.x * blockDim.x + threadIdx.x;
  if (t >= EH) return;
  const int e = t / H;
  const int h = t - e * H;
  const int sn = src[e], dn = dst[e];
  float v = el[sn * H + h] + er[dn * H + h];
  v = (v > 0.f) ? v : NEG_SLOPE * v;
  s[t] = v;
  atomicMax(&m[dn * H + h], f2ord(v));
}

// Edge pass 2: w = exp(score - max[dst]); segment-sum via atomicAdd.
template <int H>
__global__ __launch_bounds__(256) void k_edge_exp_sum(
    const int* __restrict__ dst, const int* __restrict__ m,
    float* __restrict__ s /*score in, w out*/, float* __restrict__ denom,
    int EH) {
  const int t = blockIdx.x * blockDim.x + threadIdx.x;
  if (t >= EH) return;
  const int e = t / H;
  const int h = t - e * H;
  const int dn = dst[e];
  const float w = __expf(s[t] - ord2f(m[dn * H + h]));
  s[t] = w;
  atomicAdd(&denom[dn * H + h], w);
}

// ---------------------------------------------------------------------------
// Layer-1 aggregation: out[dst, h, d] += a[e,h] * feat[src, h, d].
// One wave per edge, 8 floats per lane (float4 x2), 256 atomics/edge.
// ---------------------------------------------------------------------------
__global__ __launch_bounds__(256) void k_agg1(
    const int* __restrict__ src, const int* __restrict__ dst,
    const float* __restrict__ feat, const float* __restrict__ w,
    const float* __restrict__ denom, float* __restrict__ out, int E) {
  const int e = blockIdx.x * (blockDim.x >> 5) + (threadIdx.x >> 5);
  if (e >= E) return;
  const int lane = threadIdx.x & 31;
  const int sn = src[e], dn = dst[e];
  const int h = lane >> 2;  // 4 lanes per head
  const float a = w[e * 8 + h] / denom[dn * 8 + h];

  const float4* fs = (const float4*)(feat + (size_t)sn * KDIM);
  float4 f0 = fs[lane * 2], f1 = fs[lane * 2 + 1];
  float* op = out + (size_t)dn * KDIM + lane * 8;
  atomicAdd(op + 0, f0.x * a);
  atomicAdd(op + 1, f0.y * a);
  atomicAdd(op + 2, f0.z * a);
  atomicAdd(op + 3, f0.w * a);
  atomicAdd(op + 4, f1.x * a);
  atomicAdd(op + 5, f1.y * a);
  atomicAdd(op + 6, f1.z * a);
  atomicAdd(op + 7, f1.w * a);
}

// Layer-2 aggregation: D=40, H=1. One wave per edge, lanes cover d and d+32.
__global__ __launch_bounds__(256) void k_agg2(
    const int* __restrict__ src, const int* __restrict__ dst,
    const float* __restrict__ feat, const float* __restrict__ w,
    const float* __restrict__ denom, float* __restrict__ out, int E) {
  const int e = blockIdx.x * (blockDim.x >> 5) + (threadIdx.x >> 5);
  if (e >= E) return;
  const int lane = threadIdx.x & 31;
  const int sn = src[e], dn = dst[e];
  const float a = w[e] / denom[dn];
  atomicAdd(&out[(size_t)dn * 40 + lane], feat[(size_t)sn * 40 + lane] * a);
  if (lane < 8) {
    const int d = lane + 32;
    atomicAdd(&out[(size_t)dn * 40 + d], feat[(size_t)sn * 40 + d] * a);
  }
}

// ---------------------------------------------------------------------------
// Elementwise helpers.
// ---------------------------------------------------------------------------
__global__ __launch_bounds__(256) void k_fill_f32(float* p, int n, float v) {
  const int i = blockIdx.x * blockDim.x + threadIdx.x;
  if (i < n) p[i] = v;
}
__global__ __launch_bounds__(256) void k_fill_i32(int* p, int n, int v) {
  const int i = blockIdx.x * blockDim.x + threadIdx.x;
  if (i < n) p[i] = v;
}
// h = relu(h + b[c]), 256 columns.
__global__ __launch_bounds__(256) void k_relu_bias256(
    float* __restrict__ h, const float* __restrict__ b, int total) {
  const int i = blockIdx.x * blockDim.x + threadIdx.x;
  if (i < total) {
    const float v = h[i] + b[i & 255];
    h[i] = v > 0.f ? v : 0.f;
  }
}
// out += b[c], 40 columns.
__global__ __launch_bounds__(256) void k_bias40(
    float* __restrict__ o, const float* __restrict__ b, int total) {
  const int i = blockIdx.x * blockDim.x + threadIdx.x;
  if (i < total) o[i] += b[i % 40];
}

// ---------------------------------------------------------------------------
// Launcher.
// ---------------------------------------------------------------------------
extern "C" void kernel_launch(void* const* d_in, const int* in_sizes, int n_in,
                              void* d_out, int out_size, void* d_ws,
                              size_t ws_size, hipStream_t stream) {
  const float* X       = (const float*)d_in[0];   // [N,256]
  const int*   src     = (const int*)d_in[1];     // [E]
  const int*   dst     = (const int*)d_in[2];     // [E]
  const float* W1      = (const float*)d_in[3];   // [256,256]
  const float* attn_l1 = (const float*)d_in[4];   // [8,32]
  const float* attn_r1 = (const float*)d_in[5];   // [8,32]
  const float* b1      = (const float*)d_in[6];   // [256]
  const float* W2      = (const float*)d_in[7];   // [256,40]
  const float* attn_l2 = (const float*)d_in[8];   // [1,40]
  const float* attn_r2 = (const float*)d_in[9];   // [1,40]
  const float* b2      = (const float*)d_in[10];  // [40]
  float* out = (float*)d_out;                     // [N,40]

  const int N = in_sizes[0] / KDIM;
  const int E = in_sizes[1];

  // Workspace layout (floats).
  float* ws    = (float*)d_ws;
  float* feat1 = ws;                         // N*256
  float* h1    = feat1 + (size_t)N * KDIM;   // N*256 (agg accum -> relu'd h)
  float* feat2 = h1 + (size_t)N * KDIM;      // N*40
  float* el1   = feat2 + (size_t)N * 40;     // N*8
  float* er1   = el1 + (size_t)N * 8;        // N*8
  int*   m1    = (int*)(er1 + (size_t)N * 8);// N*8 (ordered-int max)
  float* dn1   = (float*)(m1 + (size_t)N * 8);// N*8
  float* el2   = dn1 + (size_t)N * 8;        // N
  float* er2   = el2 + N;                    // N
  int*   m2    = (int*)(er2 + N);            // N
  float* dn2   = (float*)(m2 + N);           // N
  float* w1    = dn2 + N;                    // E*8 (score then softmax weight)
  float* w2    = w1 + (size_t)E * 8;         // E

  const int TPB = 256;
  const int negInfOrd = (int)(0xFF800000u ^ 0x7FFFFFFFu);  // f2ord(-inf)
  auto blocks = [](long long n, int tpb) { return (int)((n + tpb - 1) / tpb); };

  const int Mtiles = (N + 15) / 16;

  // ---- init accumulators ----
  k_fill_i32<<<blocks(N * 8, TPB), TPB, 0, stream>>>(m1, N * 8, negInfOrd);
  k_fill_f32<<<blocks(N * 8, TPB), TPB, 0, stream>>>(dn1, N * 8, 0.f);
  k_fill_i32<<<blocks(N, TPB), TPB, 0, stream>>>(m2, N, negInfOrd);
  k_fill_f32<<<blocks(N, TPB), TPB, 0, stream>>>(dn2, N, 0.f);
  k_fill_f32<<<blocks((long long)N * KDIM, TPB), TPB, 0, stream>>>(
      h1, N * KDIM, 0.f);
  k_fill_f32<<<blocks((long long)N * 40, TPB), TPB, 0, stream>>>(
      out, N * 40, 0.f);

  // ---- layer 1 ----
  // feat1 = X @ W1  (256 cols -> 16 tiles; 8 waves/block, grid.y = 2)
  k_gemm_wmma<<<dim3(Mtiles, 2), 256, 0, stream>>>(X, W1, feat1, N, KDIM);
  k_scores1<<<blocks(N, 8), 256, 0, stream>>>(feat1, attn_l1, attn_r1, el1,
                                              er1, N);
  k_edge_score_max<8><<<blocks((long long)E * 8, TPB), TPB, 0, stream>>>(
      src, dst, el1, er1, w1, m1, E * 8);
  k_edge_exp_sum<8><<<blocks((long long)E * 8, TPB), TPB, 0, stream>>>(
      dst, m1, w1, dn1, E * 8);
  k_agg1<<<blocks(E, 8), 256, 0, stream>>>(src, dst, feat1, w1, dn1, h1, E);
  k_relu_bias256<<<blocks((long long)N * KDIM, TPB), TPB, 0, stream>>>(
      h1, b1, N * KDIM);

  // ---- layer 2 ----
  // feat2 = h1 @ W2  (40 cols -> 3 tiles; 4 waves/block, grid.y = 1)
  k_gemm_wmma<<<dim3(Mtiles, 1), 128, 0, stream>>>(h1, W2, feat2, N, 40);
  k_scores2<<<blocks(N, TPB), TPB, 0, stream>>>(feat2, attn_l2, attn_r2, el2,
                                                er2, N);
  k_edge_score_max<1><<<blocks(E, TPB), TPB, 0, stream>>>(src, dst, el2, er2,
                                                          w2, m2, E);
  k_edge_exp_sum<1><<<blocks(E, TPB), TPB, 0, stream>>>(dst, m2, w2, dn2, E);
  k_agg2<<<blocks(E, 8), 256, 0, stream>>>(src, dst, feat2, w2, dn2, out, E);
  k_bias40<<<blocks((long long)N * 40, TPB), TPB, 0, stream>>>(out, b2,
                                                               N * 40);
}